// LSS_46188078301708
// MI455X (gfx1250) — compile-verified
//
#include <hip/hip_runtime.h>

// LSS / BEV voxel pooling (scatter-add segment sum) for MI455X (gfx1250).
// Memory-bound: ~110-130 MB of traffic -> ~5us floor at 23.3 TB/s.
// No matmul structure -> WMMA not applicable; optimized via native f32
// global atomics, b128 zero-fill, NT streaming loads, wave32 blocks.

#define LSS_NX 128u
#define LSS_NY 128u
#define LSS_NZ 16u
#define LSS_C  80u

// Pass 1: zero the (B, C, NZ, NX, NY) output with 128-bit stores.
// Default (RT) temporal hint on purpose: the 84 MB output fits in the
// 192 MB global L2, so keeping the zeroed lines resident makes the
// subsequent scattered atomics L2 hits.
__global__ void lss_zero_kernel(float4* __restrict__ out, unsigned n4) {
    unsigned i = blockIdx.x * blockDim.x + threadIdx.x;
    if (i < n4) {
        out[i] = make_float4(0.f, 0.f, 0.f, 0.f);
    }
}

// Pass 2: one thread per (point, channel).
//   i = p*80 + c ; consecutive lanes -> consecutive channels of one point:
//   - x load fully coalesced (and skipped entirely for dropped points)
//   - geom loads broadcast within the wave (cache hits)
//   - unsafeAtomicAdd -> global_atomic_add_f32 (no CAS loop)
__global__ void lss_scatter_kernel(const float* __restrict__ x,
                                   const int*   __restrict__ geom,
                                   float*       __restrict__ out,
                                   unsigned total,    // npts * 80
                                   unsigned per_b) {  // points per batch
    unsigned i = blockIdx.x * blockDim.x + threadIdx.x;
    if (i >= total) return;

    unsigned p = i / LSS_C;          // point index (magic-multiply div)
    unsigned c = i - p * LSS_C;      // channel

    int gx = geom[3u * p + 0u];
    int gy = geom[3u * p + 1u];
    int gz = geom[3u * p + 2u];

    // (gx in [0,128)) & (gy in [0,128)) & (gz in [0,16)), unsigned trick
    if ((unsigned)gx < LSS_NX && (unsigned)gy < LSS_NY && (unsigned)gz < LSS_NZ) {
        unsigned b  = p / per_b;
        // output layout (B, C, NZ, NX, NY) == reference transpose(0,4,3,1,2)
        unsigned oi = ((((b * LSS_C + c) * LSS_NZ + (unsigned)gz) * LSS_NX
                        + (unsigned)gx) * LSS_NY) + (unsigned)gy;
        // Stream x through without polluting L2 (we want L2 for the output).
        float v = __builtin_nontemporal_load(&x[i]);
        unsafeAtomicAdd(&out[oi], v);
    }
}

extern "C" void kernel_launch(void* const* d_in, const int* in_sizes, int n_in,
                              void* d_out, int out_size, void* d_ws, size_t ws_size,
                              hipStream_t stream) {
    const float* x    = (const float*)d_in[0];
    const int*   geom = (const int*)d_in[1];
    float*       out  = (float*)d_out;

    const unsigned npts  = (unsigned)(in_sizes[0] / (int)LSS_C);     // B*N*D*H*W
    unsigned B = (unsigned)(out_size / (int)(LSS_C * LSS_NZ * LSS_NX * LSS_NY));
    if (B == 0u) B = 1u;
    const unsigned per_b = npts / B;

    // Pass 1: zero-fill output (out_size is a multiple of 4).
    const unsigned n4 = (unsigned)out_size / 4u;
    const unsigned zblocks = (n4 + 255u) / 256u;
    lss_zero_kernel<<<dim3(zblocks), dim3(256), 0, stream>>>((float4*)out, n4);

    // Pass 2: scatter-add (stream order guarantees zero-fill completes first).
    const unsigned total   = npts * LSS_C;                 // 37.8M < 2^31
    const unsigned sblocks = (total + 255u) / 256u;
    lss_scatter_kernel<<<dim3(sblocks), dim3(256), 0, stream>>>(
        x, geom, out, total, per_b);
}